// ContrastModule_13297218748464
// MI455X (gfx1250) — compile-verified
//
#include <hip/hip_runtime.h>
#include <hip/hip_bf16.h>
#include <math.h>

typedef __attribute__((ext_vector_type(2))) float v2f;
typedef __attribute__((ext_vector_type(8))) float v8f;

#define B_ 32
#define P_ 1024
#define L_ 32
#define H_ 128

// ---------------------------------------------------------------------------
// 1) objectness mask (argmax==1) + active count per batch
// ---------------------------------------------------------------------------
__global__ void mask_cnt_kernel(const float* __restrict__ scores,
                                float* __restrict__ mask,
                                float* __restrict__ cnt) {
  const int b = blockIdx.x;
  const int tid = threadIdx.x;
  __shared__ float red[256];
  float loc = 0.f;
  for (int p = tid; p < P_; p += 256) {
    float s0 = scores[(b * P_ + p) * 2 + 0];
    float s1 = scores[(b * P_ + p) * 2 + 1];
    float m = (s1 > s0) ? 1.f : 0.f;   // argmax ties -> index 0
    mask[b * P_ + p] = m;
    loc += m;
  }
  red[tid] = loc;
  __syncthreads();
  for (int s = 128; s > 0; s >>= 1) {
    if (tid < s) red[tid] += red[tid + s];
    __syncthreads();
  }
  if (tid == 0) cnt[b] = red[0];
}

// ---------------------------------------------------------------------------
// 2) projection Y[m][n] = sum_k X[m][k] * W[n][k]   (einsum '...d,hd->...h')
//    16-row tile per block, 8 waves cover the 128 output columns.
//    V_WMMA_F32_16X16X4_F32, K-loop of 32.
// ---------------------------------------------------------------------------
__global__ void proj_wmma_kernel(const float* __restrict__ X,
                                 const float* __restrict__ W,
                                 float* __restrict__ Y) {
  const int R    = blockIdx.x * 16;
  const int wave = threadIdx.x >> 5;
  const int lane = threadIdx.x & 31;
  const int half = lane >> 4;
  const int l2   = lane & 15;
  const int cb   = wave * 16;

  // A fragments: lane holds row R+l2, K-pair (4*kk + 2*half)
  v2f a[32];
  const float* arow = X + (size_t)(R + l2) * H_ + 2 * half;
#pragma unroll
  for (int kk = 0; kk < 32; ++kk) a[kk] = *(const v2f*)(arow + 4 * kk);

  // B fragments: B[k][n] = W[cb+n][k] -> contiguous float2 along k
  const float* wrow = W + (size_t)(cb + l2) * H_ + 2 * half;
  v8f c = {0.f, 0.f, 0.f, 0.f, 0.f, 0.f, 0.f, 0.f};
#pragma unroll
  for (int kk = 0; kk < 32; ++kk) {
    v2f bf = *(const v2f*)(wrow + 4 * kk);
    c = __builtin_amdgcn_wmma_f32_16x16x4_f32(false, a[kk], false, bf,
                                              (short)0, c, false, false);
  }
  // C/D layout: VGPR v holds row v+8*half, col = lane&15
#pragma unroll
  for (int v = 0; v < 8; ++v)
    Y[(size_t)(R + v + 8 * half) * H_ + cb + l2] = c[v];
}

// ---------------------------------------------------------------------------
// 3) row L2-normalize in place (eps = 1e-12, matches F.normalize clip)
// ---------------------------------------------------------------------------
__global__ void normalize_kernel(float* __restrict__ Y) {
  const int row = blockIdx.x;
  const int tid = threadIdx.x;   // 128
  __shared__ float red[128];
  float x = Y[(size_t)row * H_ + tid];
  red[tid] = x * x;
  __syncthreads();
  for (int s = 64; s > 0; s >>= 1) {
    if (tid < s) red[tid] += red[tid + s];
    __syncthreads();
  }
  float d = fmaxf(sqrtf(red[0]), 1e-12f);
  Y[(size_t)row * H_ + tid] = x / d;
}

// ---------------------------------------------------------------------------
// 4) tgt[b,l,p] = (AABB IoU(gt_l (+0.01 size), pred_p) > 0.25) * mask[p]
// ---------------------------------------------------------------------------
__global__ void tgt_kernel(const float* __restrict__ gc, const float* __restrict__ gs,
                           const float* __restrict__ pc, const float* __restrict__ ps,
                           const float* __restrict__ mask, float* __restrict__ tgt) {
  const int idx = blockIdx.x * blockDim.x + threadIdx.x;  // b*L*P + l*P + p
  const int b = idx >> 15;          // L_*P_ = 32768
  const int rem = idx & 32767;
  const int l = rem >> 10;          // P_ = 1024
  const int p = rem & 1023;
  float inter = 1.f, vg = 1.f, vp = 1.f;
#pragma unroll
  for (int d = 0; d < 3; ++d) {
    float gcv = gc[(b * L_ + l) * 3 + d];
    float gsv = gs[(b * L_ + l) * 3 + d] + 0.01f;
    float pcv = pc[(b * P_ + p) * 3 + d];
    float psv = ps[(b * P_ + p) * 3 + d];
    float gmin = gcv - 0.5f * gsv, gmax = gcv + 0.5f * gsv;
    float pmin = pcv - 0.5f * psv, pmax = pcv + 0.5f * psv;
    float ov = fminf(gmax, pmax) - fmaxf(gmin, pmin);
    inter *= fmaxf(ov, 0.f);
    vg *= gsv;
    vp *= psv;
  }
  float iou = inter / (vg + vp - inter + 1e-7f);
  tgt[idx] = (iou > 0.25f) ? mask[b * P_ + p] : 0.f;
}

// ---------------------------------------------------------------------------
// 5) Gram matrix S = boxi . boxi^T, fused streaming row logsumexp (shift 1.0):
//    lse[b,p] = 1 + log( sum_{q active} exp(S[p,q] - 1) )
//    One block per (16-row tile, batch); 8 waves stripe the 64 column tiles.
// ---------------------------------------------------------------------------
__global__ void gram_lse_kernel(const float* __restrict__ boxi,
                                const float* __restrict__ mask,
                                float* __restrict__ lse) {
  const int I    = blockIdx.x;
  const int b    = blockIdx.y;
  const int R    = I * 16;
  const int wave = threadIdx.x >> 5;
  const int lane = threadIdx.x & 31;
  const int half = lane >> 4;
  const int l2   = lane & 15;
  const float* X = boxi + (size_t)b * P_ * H_;

  __shared__ float rowsum[16];
  if (threadIdx.x < 16) rowsum[threadIdx.x] = 0.f;
  __syncthreads();

  v2f a[32];
  const float* arow = X + (size_t)(R + l2) * H_ + 2 * half;
#pragma unroll
  for (int kk = 0; kk < 32; ++kk) a[kk] = *(const v2f*)(arow + 4 * kk);

  float es[8] = {0.f, 0.f, 0.f, 0.f, 0.f, 0.f, 0.f, 0.f};

  for (int Jt = wave; Jt < P_ / 16; Jt += 8) {
    const int C0 = Jt * 16;
    const float* brow = X + (size_t)(C0 + l2) * H_ + 2 * half;
    const float mcol = mask[b * P_ + C0 + l2];   // column mask for this lane
    v8f c = {0.f, 0.f, 0.f, 0.f, 0.f, 0.f, 0.f, 0.f};
#pragma unroll
    for (int kk = 0; kk < 32; ++kk) {
      v2f bf = *(const v2f*)(brow + 4 * kk);
      c = __builtin_amdgcn_wmma_f32_16x16x4_f32(false, a[kk], false, bf,
                                                (short)0, c, false, false);
    }
#pragma unroll
    for (int v = 0; v < 8; ++v) es[v] += mcol * __expf(c[v] - 1.0f);
  }
#pragma unroll
  for (int v = 0; v < 8; ++v) atomicAdd(&rowsum[v + 8 * half], es[v]);
  __syncthreads();
  if (threadIdx.x < 16)
    lse[b * P_ + R + threadIdx.x] = 1.0f + __logf(rowsum[threadIdx.x]);
}

// ---------------------------------------------------------------------------
// 6) lang part per (b,l): lse over active p of text.boxl, tgt-weighted dot,
//    tgt count s_l
// ---------------------------------------------------------------------------
__global__ void lang_kernel(const float* __restrict__ text,
                            const float* __restrict__ boxl,
                            const float* __restrict__ mask,
                            const float* __restrict__ tgt,
                            float* __restrict__ lsel,
                            float* __restrict__ dotl,
                            float* __restrict__ sl) {
  const int idx = blockIdx.x;     // b*L + l
  const int b = idx >> 5;         // L_ = 32
  const int tid = threadIdx.x;    // 256
  __shared__ float tl[H_];
  __shared__ float r1[256], r2[256], r3[256];
  if (tid < H_) tl[tid] = text[(size_t)idx * H_ + tid];
  __syncthreads();
  float es = 0.f, ds = 0.f, ss = 0.f;
  for (int i = 0; i < 4; ++i) {
    const int p = tid + i * 256;
    const float* bp = boxl + ((size_t)b * P_ + p) * H_;
    float s = 0.f;
#pragma unroll 8
    for (int h = 0; h < H_; ++h) s = fmaf(tl[h], bp[h], s);
    const float m = mask[b * P_ + p];
    const float t = tgt[(size_t)idx * P_ + p];
    es += m * __expf(s - 1.0f);
    ds += t * s;
    ss += t;
  }
  r1[tid] = es; r2[tid] = ds; r3[tid] = ss;
  __syncthreads();
  for (int s = 128; s > 0; s >>= 1) {
    if (tid < s) { r1[tid] += r1[tid + s]; r2[tid] += r2[tid + s]; r3[tid] += r3[tid + s]; }
    __syncthreads();
  }
  if (tid == 0) { lsel[idx] = 1.0f + __logf(r1[0]); dotl[idx] = r2[0]; sl[idx] = r3[0]; }
}

// ---------------------------------------------------------------------------
// 7) per (b,l): ||sum_p tgt*boxi[p]||^2  and  a_l = sum_p tgt*lse_iou[p]
// ---------------------------------------------------------------------------
__global__ void va_kernel(const float* __restrict__ boxi,
                          const float* __restrict__ tgt,
                          const float* __restrict__ lsei,
                          float* __restrict__ vn,
                          float* __restrict__ al) {
  const int idx = blockIdx.x;    // b*L + l
  const int b = idx >> 5;
  const int tid = threadIdx.x;   // 128 = H_
  const float* trow = tgt + (size_t)idx * P_;
  float v = 0.f;
  for (int p = 0; p < P_; ++p) {
    const float t = trow[p];     // uniform across block -> no divergence
    if (t != 0.f) v += t * boxi[((size_t)b * P_ + p) * H_ + tid];
  }
  float a = 0.f;
  for (int p = tid; p < P_; p += 128) {
    const float t = trow[p];
    if (t != 0.f) a += t * lsei[b * P_ + p];
  }
  __shared__ float r1[128], r2[128];
  r1[tid] = v * v; r2[tid] = a;
  __syncthreads();
  for (int s = 64; s > 0; s >>= 1) {
    if (tid < s) { r1[tid] += r1[tid + s]; r2[tid] += r2[tid + s]; }
    __syncthreads();
  }
  if (tid == 0) { vn[idx] = r1[0]; al[idx] = r2[0]; }
}

// ---------------------------------------------------------------------------
// 8) final scalar losses
//    lang: 0.5*(s*lse - dot)/c1 ; iou: (a*s - ||v||^2)/c1^2 ; masked, /B
// ---------------------------------------------------------------------------
__global__ void final_kernel(const float* __restrict__ cnt,
                             const float* __restrict__ sl,
                             const float* __restrict__ lsel,
                             const float* __restrict__ dotl,
                             const float* __restrict__ al,
                             const float* __restrict__ vn,
                             const int* __restrict__ lang_num,
                             float* __restrict__ out) {
  const int tid = threadIdx.x;
  __shared__ float r1[256], r2[256];
  float lsum = 0.f, isum = 0.f;
  for (int idx = tid; idx < B_ * L_; idx += 256) {
    const int b = idx >> 5;
    const int l = idx & 31;
    const float act = (l < lang_num[b]) ? 1.f : 0.f;
    const float c1 = fmaxf(cnt[b], 1.f);
    const float s = sl[idx];
    float lossv = 0.f, lv = 0.f;
    if (s > 0.f) {   // guards 0 * (-inf) when no active columns / matches
      lossv = (s * lsel[idx] - dotl[idx]) / c1;
      lv = (al[idx] * s - vn[idx]) / (c1 * c1);
    }
    lsum += act * 0.5f * lossv;
    isum += act * lv;
  }
  r1[tid] = lsum; r2[tid] = isum;
  __syncthreads();
  for (int s = 128; s > 0; s >>= 1) {
    if (tid < s) { r1[tid] += r1[tid + s]; r2[tid] += r2[tid + s]; }
    __syncthreads();
  }
  if (tid == 0) { out[0] = r1[0] / (float)B_; out[1] = r2[0] / (float)B_; }
}

// ---------------------------------------------------------------------------
extern "C" void kernel_launch(void* const* d_in, const int* in_sizes, int n_in,
                              void* d_out, int out_size, void* d_ws, size_t ws_size,
                              hipStream_t stream) {
  const float* pred_center = (const float*)d_in[0];
  const float* pred_size   = (const float*)d_in[1];
  const float* bbox        = (const float*)d_in[2];
  const float* gt_center   = (const float*)d_in[3];
  const float* gt_size     = (const float*)d_in[4];
  const float* lang_emb    = (const float*)d_in[5];
  const float* scores      = (const float*)d_in[6];
  const float* Wt          = (const float*)d_in[7];
  const float* Wp          = (const float*)d_in[8];
  const float* Wpi         = (const float*)d_in[9];
  const int*   lang_num    = (const int*)d_in[10];
  float* out = (float*)d_out;

  float* ws = (float*)d_ws;
  size_t o = 0;
  float* mask = ws + o; o += (size_t)B_ * P_;
  float* cnt  = ws + o; o += B_;
  float* text = ws + o; o += (size_t)B_ * L_ * H_;
  float* boxl = ws + o; o += (size_t)B_ * P_ * H_;
  float* boxi = ws + o; o += (size_t)B_ * P_ * H_;
  float* tgt  = ws + o; o += (size_t)B_ * L_ * P_;
  float* lsel = ws + o; o += (size_t)B_ * L_;
  float* dotl = ws + o; o += (size_t)B_ * L_;
  float* sl   = ws + o; o += (size_t)B_ * L_;
  float* lsei = ws + o; o += (size_t)B_ * P_;
  float* vn   = ws + o; o += (size_t)B_ * L_;
  float* al   = ws + o; o += (size_t)B_ * L_;
  (void)in_sizes; (void)n_in; (void)out_size; (void)ws_size;

  mask_cnt_kernel<<<B_, 256, 0, stream>>>(scores, mask, cnt);
  proj_wmma_kernel<<<(B_ * P_) / 16, 256, 0, stream>>>(bbox, Wp, boxl);
  proj_wmma_kernel<<<(B_ * P_) / 16, 256, 0, stream>>>(bbox, Wpi, boxi);
  proj_wmma_kernel<<<(B_ * L_) / 16, 256, 0, stream>>>(lang_emb, Wt, text);
  normalize_kernel<<<B_ * P_, 128, 0, stream>>>(boxl);
  normalize_kernel<<<B_ * P_, 128, 0, stream>>>(boxi);
  normalize_kernel<<<B_ * L_, 128, 0, stream>>>(text);
  tgt_kernel<<<(B_ * L_ * P_) / 256, 256, 0, stream>>>(gt_center, gt_size,
                                                       pred_center, pred_size,
                                                       mask, tgt);
  gram_lse_kernel<<<dim3(P_ / 16, B_), 256, 0, stream>>>(boxi, mask, lsei);
  lang_kernel<<<B_ * L_, 256, 0, stream>>>(text, boxl, mask, tgt, lsel, dotl, sl);
  va_kernel<<<B_ * L_, 128, 0, stream>>>(boxi, tgt, lsei, vn, al);
  final_kernel<<<1, 256, 0, stream>>>(cnt, sl, lsel, dotl, al, vn, lang_num, out);
}